// LinearChainCrfDependency_41515153883242
// MI455X (gfx1250) — compile-verified
//
#include <hip/hip_runtime.h>
#include <hip/hip_bf16.h>

// Problem constants from the reference
#define BB 256
#define TT 2048
#define KK 64

typedef __attribute__((ext_vector_type(2))) float v2f;
typedef __attribute__((ext_vector_type(8))) float v8f;

#define ESTRIDE 66   // padded LDS row stride for E = exp(trans) (64x64)
#define ASTRIDE 68   // padded LDS row stride for alpha / expS (16x64)

// ---------------------------------------------------------------------------
// Forward recursion: alpha scan over T with WMMA f32 16x16x4 log-space matmul.
// grid = B/16 blocks, 128 threads (4 waves); wave w owns j-tile [16w,16w+16).
// K=64 reduction split across 4 independent WMMA accumulator chains (depth 4
// instead of 16) since each SIMD runs a single wave and has no cross-wave
// latency hiding.
// ---------------------------------------------------------------------------
__global__ __launch_bounds__(128)
void crf_forward_kernel(const float* __restrict__ em,
                        const unsigned char* __restrict__ mask, // jax bool = 1 byte
                        const float* __restrict__ start,
                        const float* __restrict__ endv,
                        const float* __restrict__ trans,
                        float* __restrict__ logZ)
{
    __shared__ float Elds[64 * ESTRIDE];
    __shared__ float alphaS[16 * ASTRIDE];
    __shared__ float expS[16 * ASTRIDE];
    __shared__ float mxS[16];
    __shared__ float endS[64];
    __shared__ unsigned char mkS[16];

    const int tid  = threadIdx.x;
    const int lane = tid & 31;
    const int wid  = tid >> 5;       // wave id 0..3
    const int half = lane >> 4;      // 0/1 (lane half selects K sub-pair)
    const int l16  = lane & 15;
    const int j0   = wid * 16;
    const int rowbase = blockIdx.x * 16;

    // E = exp(trans), computed once per block into LDS
    for (int idx = tid; idx < 64 * 64; idx += 128) {
        int i = idx >> 6, j = idx & 63;
        Elds[i * ESTRIDE + j] = __expf(trans[idx]);
    }
    // alpha0 = start + emissions[:,0,:]
    for (int idx = tid; idx < 16 * 64; idx += 128) {
        int m = idx >> 6, k = idx & 63;
        alphaS[m * ASTRIDE + k] = start[k] + em[(size_t)(rowbase + m) * TT * KK + k];
    }
    if (tid < 64) endS[tid] = endv[tid];
    __syncthreads();

    // B fragments of E for this wave's j-tile: constant across all timesteps.
    // B(4x16) layout: V0 = E[k0][n], V1 = E[k0+1][n], k0 = 4c + 2*half, n = j0+l16
    v2f bf[16];
#pragma unroll
    for (int c = 0; c < 16; ++c) {
        int k0 = 4 * c + 2 * half;
        int n  = j0 + l16;
        bf[c].x = Elds[k0 * ESTRIDE + n];
        bf[c].y = Elds[(k0 + 1) * ESTRIDE + n];
    }

    const int row = tid >> 3;   // 0..15 (batch row within tile)
    const int sub = tid & 7;    // 0..7

    // Per-thread epilogue addressing: row M=8*half (r=0) element for this lane's column.
    const size_t embase0 = (size_t)(rowbase + 8 * half) * TT * KK + (j0 + l16);

    for (int t = 1; t < TT; ++t) {
        // ---- phase 1: per-row max of alpha (8 lanes per row + shfl butterfly)
        float mv = -3.4e38f;
#pragma unroll
        for (int i = 0; i < 8; ++i)
            mv = fmaxf(mv, alphaS[row * ASTRIDE + sub + 8 * i]);
        mv = fmaxf(mv, __shfl_xor(mv, 1));
        mv = fmaxf(mv, __shfl_xor(mv, 2));
        mv = fmaxf(mv, __shfl_xor(mv, 4));
        if (sub == 0) mxS[row] = mv;
        __syncthreads();

        // ---- phase 2: expS = exp(alpha - mx), shared by all 4 waves
        {
            float mrow = mxS[row];
            int kb = sub * 8;
#pragma unroll
            for (int i = 0; i < 8; ++i)
                expS[row * ASTRIDE + kb + i] =
                    __expf(alphaS[row * ASTRIDE + kb + i] - mrow);
        }
        if (tid < 16) mkS[tid] = mask[(size_t)(rowbase + tid) * TT + t];
        __syncthreads();

        // ---- prefetch this step's emission tile (independent of WMMA chain)
        float emr[8];
#pragma unroll
        for (int r = 0; r < 8; ++r)
            emr[r] = em[embase0 + (size_t)r * TT * KK + (size_t)t * KK];

        // ---- phase 3: s = exp(alpha-m) @ E via 4 independent WMMA chains
        v8f acc[4];
#pragma unroll
        for (int q = 0; q < 4; ++q)
            acc[q] = (v8f){0.f, 0.f, 0.f, 0.f, 0.f, 0.f, 0.f, 0.f};
#pragma unroll
        for (int c = 0; c < 16; ++c) {
            int k0 = 4 * c + 2 * half;
            v2f a;
            a.x = expS[l16 * ASTRIDE + k0];
            a.y = expS[l16 * ASTRIDE + k0 + 1];
            acc[c & 3] = __builtin_amdgcn_wmma_f32_16x16x4_f32(
                false, a, false, bf[c], (short)0, acc[c & 3], false, false);
        }

        // ---- phase 4: new = log(s) + mx + em[:,t,:], masked blend
#pragma unroll
        for (int r = 0; r < 8; ++r) {
            int M = r + 8 * half;            // C/D layout: lane half selects M+8
            int n = j0 + l16;
            float sv = (acc[0][r] + acc[1][r]) + (acc[2][r] + acc[3][r]);
            float cand = __logf(sv) + mxS[M] + emr[r];
            float oldv = alphaS[M * ASTRIDE + n];
            alphaS[M * ASTRIDE + n] = mkS[M] ? cand : oldv;
        }
        __syncthreads();   // alpha complete before next iteration reads
    }

    // ---- logZ[b] = LSE_k(alpha[b,k] + end[k])
    {
        float mx2 = -3.4e38f;
#pragma unroll
        for (int i = 0; i < 8; ++i) {
            int k = sub + 8 * i;
            mx2 = fmaxf(mx2, alphaS[row * ASTRIDE + k] + endS[k]);
        }
        mx2 = fmaxf(mx2, __shfl_xor(mx2, 1));
        mx2 = fmaxf(mx2, __shfl_xor(mx2, 2));
        mx2 = fmaxf(mx2, __shfl_xor(mx2, 4));
        float s = 0.f;
#pragma unroll
        for (int i = 0; i < 8; ++i) {
            int k = sub + 8 * i;
            s += __expf(alphaS[row * ASTRIDE + k] + endS[k] - mx2);
        }
        s += __shfl_xor(s, 1);
        s += __shfl_xor(s, 2);
        s += __shfl_xor(s, 4);
        if (sub == 0) logZ[rowbase + row] = __logf(s) + mx2;
    }
}

// ---------------------------------------------------------------------------
// Gold path score per batch row (gather + reduction). One block per b.
// ---------------------------------------------------------------------------
__global__ __launch_bounds__(256)
void crf_gold_kernel(const float* __restrict__ em,
                     const int* __restrict__ tags,
                     const unsigned char* __restrict__ mask,
                     const float* __restrict__ start,
                     const float* __restrict__ endv,
                     const float* __restrict__ trans,
                     float* __restrict__ goldw)
{
    __shared__ float sred[256];
    __shared__ int   ired[256];
    const int b   = blockIdx.x;
    const int tid = threadIdx.x;
    const size_t base = (size_t)b * TT;

    int plen = 0;
    for (int t = tid; t < TT; t += 256)
        plen += mask[base + t] ? 1 : 0;

    float psum = 0.f;
    for (int p = tid; p < TT - 1; p += 256) {
        if (mask[base + p] && mask[base + p + 1]) {
            int pv = tags[base + p];
            int cv = tags[base + p + 1];
            psum += trans[pv * KK + cv] + em[(base + p + 1) * KK + cv];
        }
    }
    sred[tid] = psum;
    ired[tid] = plen;
    __syncthreads();
    for (int s = 128; s > 0; s >>= 1) {
        if (tid < s) { sred[tid] += sred[tid + s]; ired[tid] += ired[tid + s]; }
        __syncthreads();
    }
    if (tid == 0) {
        int t0 = tags[base];
        float sc = (start[t0] + em[base * KK + t0]) * (mask[base] ? 1.f : 0.f);
        int last = ired[0] - 1;        // lengths >= 1 per reference
        sc += sred[0] + endv[tags[base + last]];
        goldw[b] = sc;
    }
}

// ---------------------------------------------------------------------------
// out = mean(logZ - gold)
// ---------------------------------------------------------------------------
__global__ __launch_bounds__(256)
void crf_mean_kernel(const float* __restrict__ logZ,
                     const float* __restrict__ goldw,
                     float* __restrict__ out)
{
    __shared__ float sred[256];
    int tid = threadIdx.x;
    sred[tid] = logZ[tid] - goldw[tid];
    __syncthreads();
    for (int s = 128; s > 0; s >>= 1) {
        if (tid < s) sred[tid] += sred[tid + s];
        __syncthreads();
    }
    if (tid == 0) out[0] = sred[0] / (float)BB;
}

extern "C" void kernel_launch(void* const* d_in, const int* in_sizes, int n_in,
                              void* d_out, int out_size, void* d_ws, size_t ws_size,
                              hipStream_t stream)
{
    // setup_inputs order: emissions(f32), tags(i32), mask(bool/1B), start, end, trans
    const float* em            = (const float*)d_in[0];
    const int* tags            = (const int*)d_in[1];
    const unsigned char* mask  = (const unsigned char*)d_in[2];
    const float* start         = (const float*)d_in[3];
    const float* endv          = (const float*)d_in[4];
    const float* trans         = (const float*)d_in[5];
    float* out = (float*)d_out;

    float* logZ  = (float*)d_ws;       // 256 floats
    float* goldw = logZ + BB;          // 256 floats

    crf_forward_kernel<<<BB / 16, 128, 0, stream>>>(em, mask, start, endv, trans, logZ);
    crf_gold_kernel<<<BB, 256, 0, stream>>>(em, tags, mask, start, endv, trans, goldw);
    crf_mean_kernel<<<1, 256, 0, stream>>>(logZ, goldw, out);
}